// ContrastiveReact_25297357373524
// MI455X (gfx1250) — compile-verified
//
#include <hip/hip_runtime.h>
#include <math.h>
#include <stdint.h>

typedef __attribute__((ext_vector_type(2))) float v2f;
typedef __attribute__((ext_vector_type(4))) float v4f;
typedef __attribute__((ext_vector_type(8))) float v8f;
typedef __attribute__((ext_vector_type(4))) unsigned int v4u;
typedef __attribute__((ext_vector_type(4))) int v4i;
typedef __attribute__((ext_vector_type(8))) int v8i;

#define DIM      128
#define BATCH    32
#define NPOS     16
#define NNEG     128
#define NGROUP   8
#define NPURCH   50000
#define NPOSQ    (BATCH * NPOS)            // 512
#define NNEGQ    (BATCH * NNEG)            // 4096
#define NQ       (NPOSQ + NNEGQ)           // 4608
#define PCHUNKS  (NPURCH / 16)             // 3125 (exact)
#define NWAVES   8
#define QTILES   (NQ / 16)                 // 288
#define NSLICE   8                         // P-dimension grid split
#define QGROUPS  (QTILES / NWAVES)         // 36
#define PITCH    132                       // 128 dwords + 4 pad dwords (bank-conflict free)
#define CHUNKBYTES (16 * DIM * 4)          // 8 KB per 16-row purch chunk

#define HAVE_TDM __has_builtin(__builtin_amdgcn_tensor_load_to_lds)

// ---------------------------------------------------------------------------
// 1/||row|| for a [rows x 128] f32 matrix, one thread per row (float4 loads).
// ---------------------------------------------------------------------------
__global__ void inv_norm_kernel(const float* __restrict__ X,
                                float* __restrict__ out, int rows) {
    int r = blockIdx.x * blockDim.x + threadIdx.x;
    if (r >= rows) return;
    const float4* p = (const float4*)(X + (size_t)r * DIM);
    float s = 0.0f;
#pragma unroll
    for (int i = 0; i < DIM / 4; ++i) {
        float4 v = p[i];
        s += v.x * v.x + v.y * v.y + v.z * v.z + v.w * v.w;
    }
    out[r] = 1.0f / fmaxf(sqrtf(s), 1e-20f);
}

#if HAVE_TDM
// Issue a TDM load of one 16x128 f32 purch chunk into LDS with a 4-dword pad
// after every 128 dwords (row pitch 132 dwords). D# packed per ISA 8.3/8.4.
// This toolchain's builtin takes 6 args (g0, g1, g2, g3, extra group, cpol).
__device__ __forceinline__ void tdm_issue(const float* purch, int ch,
                                          unsigned lds_off) {
    unsigned long long ga =
        (unsigned long long)(uintptr_t)purch + (unsigned long long)ch * CHUNKBYTES;
    v4u g0;
    g0.x = 1u;                                   // count=1 (valid descriptor)
    g0.y = lds_off;                              // lds_addr (bytes)
    g0.z = (unsigned)ga;                         // global_addr[31:0]
    g0.w = (unsigned)((ga >> 32) & 0x01FFFFFFu)  // global_addr[56:32]
           | 0x80000000u;                        // type=2 ("image")
    v8i g1;
    g1[0] = (2 << 16)      // data_size = 4 bytes
          | (1 << 20)      // pad_enable
          | (6 << 22)      // pad_interval: 128 dwords
          | (3 << 25);     // pad_amount: 4 dwords
    g1[1] = (DIM & 0xFFFF) << 16;          // tensor_dim0 = 128 (bits 79:48 lo)
    g1[2] = (int)((unsigned)(NPURCH & 0xFFFF) << 16);  // tensor_dim1 lo16
    g1[3] = (DIM & 0xFFFF) << 16;          // tensor_dim1 hi16=0 | tile_dim0 = 128
    g1[4] = 16;                            // tile_dim1 = 16 rows | tile_dim2 = 0
    g1[5] = DIM;                           // tensor_dim0_stride = 128 (lo32)
    g1[6] = 0;
    g1[7] = 0;
    v4i z4 = {};
    v8i z8 = {};
    __builtin_amdgcn_tensor_load_to_lds(g0, g1, z4, z4, z8, 0);
}
#endif

// ---------------------------------------------------------------------------
// Block = 8 waves sharing a TDM-staged purch chunk in LDS; wave w computes a
// different 16-query tile vs the same chunk via V_WMMA_F32_16X16X4_F32.
// Grid: (NSLICE, QGROUPS). Writes per-slice partial column maxes.
// ---------------------------------------------------------------------------
__global__ __launch_bounds__(256) void maxsim_kernel(
    const float* __restrict__ purch, const float* __restrict__ pos,
    const float* __restrict__ neg,   const float* __restrict__ invP,
    float* __restrict__ partial) {
    __shared__ float lbuf[2][16 * PITCH];   // 2 x 8448 B, padded pitch

    const int tid  = threadIdx.x;
    const int lane = tid & 31;
    const int wave = tid >> 5;
    const int nm   = lane & 15;          // A: row M; B/C: column N
    const int koff = (lane >> 4) * 2;    // which K pair within a K=4 step

    const int slice = blockIdx.x;                // 0..7: strided P chunks
    const int tile  = blockIdx.y * NWAVES + wave;// 0..287: this wave's queries

    const float* qbase = (tile < NPOSQ / 16)
        ? (pos + (size_t)tile * 16 * DIM)
        : (neg + (size_t)(tile - NPOSQ / 16) * 16 * DIM);

    // Preload A (16x128 query tile) into registers in WMMA A-layout.
    v2f a[32];
    const float* qr = qbase + (size_t)nm * DIM + koff;
#pragma unroll
    for (int kk = 0; kk < 32; ++kk)
        a[kk] = *(const v2f*)(qr + kk * 4);

    float rmax[8];
#pragma unroll
    for (int r = 0; r < 8; ++r) rmax[r] = -3.4e38f;

    const int nchunks = (PCHUNKS - slice + NSLICE - 1) / NSLICE;

#if HAVE_TDM
    if (wave == 0) tdm_issue(purch, slice, (unsigned)(uintptr_t)(void*)&lbuf[0][0]);
#endif

    for (int i = 0; i < nchunks; ++i) {
        const int ch = slice + NSLICE * i;
#if HAVE_TDM
        const int buf = i & 1;
        if (wave == 0) __builtin_amdgcn_s_wait_tensorcnt((short)0);
        __syncthreads();                 // chunk i staged & visible to block
        if (wave == 0 && (i + 1) < nchunks)
            tdm_issue(purch, slice + NSLICE * (i + 1),
                      (unsigned)(uintptr_t)(void*)&lbuf[buf ^ 1][0]);
#else
        const int buf = 0;
        // Cooperative coalesced staging: 512 x 16B segments over 256 threads.
        {
            const v4f* src = (const v4f*)(purch + (size_t)ch * 16 * DIM);
#pragma unroll
            for (int t = 0; t < 2; ++t) {
                const int s   = tid * 2 + t;
                const int row = s >> 5, c4 = s & 31;
                *(v4f*)&lbuf[0][row * PITCH + c4 * 4] = src[row * 32 + c4];
            }
        }
        __syncthreads();
#endif
        // Phase 1: pull this wave's whole B fragment (16 x b128, one clause;
        // conflict-free: lane n -> bank quad 4n). Phase 2: 32 WMMAs in one
        // burst over 4 independent accumulator chains (dep distance 4).
        const float* lb = &lbuf[buf][0] + nm * PITCH + koff;
        v4f bb[16];
#pragma unroll
        for (int j = 0; j < 16; ++j)
            bb[j] = *(const v4f*)(lb + j * 8);

        v8f c0 = {}, c1 = {}, c2 = {}, c3 = {};
#pragma unroll
        for (int j = 0; j < 16; j += 2) {
            v2f b0; b0.x = bb[j].x;     b0.y = bb[j].y;
            v2f b1; b1.x = bb[j].z;     b1.y = bb[j].w;
            v2f b2; b2.x = bb[j + 1].x; b2.y = bb[j + 1].y;
            v2f b3; b3.x = bb[j + 1].z; b3.y = bb[j + 1].w;
            c0 = __builtin_amdgcn_wmma_f32_16x16x4_f32(
                     false, a[2 * j],     false, b0, (short)0, c0, false, false);
            c1 = __builtin_amdgcn_wmma_f32_16x16x4_f32(
                     false, a[2 * j + 1], false, b1, (short)0, c1, false, false);
            c2 = __builtin_amdgcn_wmma_f32_16x16x4_f32(
                     false, a[2 * j + 2], false, b2, (short)0, c2, false, false);
            c3 = __builtin_amdgcn_wmma_f32_16x16x4_f32(
                     false, a[2 * j + 3], false, b3, (short)0, c3, false, false);
        }
        const float inp = invP[ch * 16 + nm];
#pragma unroll
        for (int r = 0; r < 8; ++r)
            rmax[r] = fmaxf(rmax[r], ((c0[r] + c1[r]) + (c2[r] + c3[r])) * inp);
        __syncthreads();                 // all waves done reading buf
    }

    // C layout: lanes 0-15 hold rows 0-7, lanes 16-31 rows 8-15; max across
    // the 16 lanes of each half, then two lanes publish this wave's 16 rows.
#pragma unroll
    for (int r = 0; r < 8; ++r) {
        float v = rmax[r];
#pragma unroll
        for (int m = 1; m < 16; m <<= 1)
            v = fmaxf(v, __shfl_xor(v, m, 32));
        rmax[r] = v;
    }
    float* dst = partial + (size_t)slice * NQ + tile * 16;
    if (lane == 0) {
#pragma unroll
        for (int r = 0; r < 8; ++r) dst[r] = rmax[r];
    } else if (lane == 16) {
#pragma unroll
        for (int r = 0; r < 8; ++r) dst[8 + r] = rmax[r];
    }
}

// ---------------------------------------------------------------------------
// Combine per-slice maxes: dist[q] = 1 - invQ[q] * max_s partial[s][q].
// ---------------------------------------------------------------------------
__global__ void combine_kernel(const float* __restrict__ partial,
                               const float* __restrict__ invQ,
                               float* __restrict__ dist) {
    int q = blockIdx.x * blockDim.x + threadIdx.x;
    if (q >= NQ) return;
    float v = partial[q];
#pragma unroll
    for (int s = 1; s < NSLICE; ++s) v = fmaxf(v, partial[(size_t)s * NQ + q]);
    dist[q] = 1.0f - invQ[q] * v;
}

// ---------------------------------------------------------------------------
// Tiny finalization: one wave, one lane per batch element.
// ---------------------------------------------------------------------------
__global__ void loss_kernel(const float* __restrict__ dist,
                            const float* __restrict__ cost_pos,
                            const float* __restrict__ cost_neg,
                            const int* __restrict__ seg,
                            float* __restrict__ out) {
    __shared__ float ss[BATCH][NGROUP];
    const int b = threadIdx.x;  // 0..31
#pragma unroll
    for (int g = 0; g < NGROUP; ++g) ss[b][g] = 0.0f;

    const float* dp = dist + b * NPOS;
    float pv = 0.0f;
#pragma unroll
    for (int j = 0; j < NPOS; ++j) pv += dp[j];
    pv = pv + cost_pos[b];  // MULT=1, TEMPERATURE=1

    const float* dn = dist + NPOSQ + b * NNEG;
    const int*   sb = seg + b * NNEG;
    for (int j = 0; j < NNEG; ++j) ss[b][sb[j]] += dn[j];

    float den = 0.0f;
#pragma unroll
    for (int g = 0; g < NGROUP; ++g)
        den += expf(-(ss[b][g] + cost_neg[b * NGROUP + g]));

    const float num = expf(-pv);
    float loss = logf(num + den) + pv;   // == -log(num/(num+den))

#pragma unroll
    for (int m = 1; m < 32; m <<= 1)
        loss += __shfl_xor(loss, m, 32);
    if (b == 0) out[0] = loss * (1.0f / BATCH);
}

// ---------------------------------------------------------------------------
extern "C" void kernel_launch(void* const* d_in, const int* in_sizes, int n_in,
                              void* d_out, int out_size, void* d_ws, size_t ws_size,
                              hipStream_t stream) {
    (void)in_sizes; (void)n_in; (void)out_size; (void)ws_size;
    const float* purch    = (const float*)d_in[0];  // [50000,128]
    const float* pos      = (const float*)d_in[1];  // [32,16,128]
    const float* neg      = (const float*)d_in[2];  // [32,128,128]
    const float* cost_pos = (const float*)d_in[3];  // [32]
    const float* cost_neg = (const float*)d_in[4];  // [32,8]
    const int*   seg      = (const int*)d_in[5];    // [32,128]
    float* out = (float*)d_out;

    float* invP    = (float*)d_ws;            // 50000
    float* invQ    = invP + NPURCH;           // 4608 (pos rows, then neg rows)
    float* dist    = invQ + NQ;               // 4608
    float* partial = dist + NQ;               // 8 * 4608

    inv_norm_kernel<<<(NPURCH + 255) / 256, 256, 0, stream>>>(purch, invP, NPURCH);
    inv_norm_kernel<<<(NPOSQ  + 255) / 256, 256, 0, stream>>>(pos,   invQ, NPOSQ);
    inv_norm_kernel<<<(NNEGQ  + 255) / 256, 256, 0, stream>>>(neg,   invQ + NPOSQ, NNEGQ);

    maxsim_kernel<<<dim3(NSLICE, QGROUPS), 256, 0, stream>>>(purch, pos, neg,
                                                             invP, partial);

    combine_kernel<<<(NQ + 255) / 256, 256, 0, stream>>>(partial, invQ, dist);

    loss_kernel<<<1, 32, 0, stream>>>(dist, cost_pos, cost_neg, seg, out);
}